// SelectiveScanStateWithInitState_5136780886275
// MI455X (gfx1250) — compile-verified
//
#include <hip/hip_runtime.h>
#include <stdint.h>

// Problem constants (match reference): B_SZ=2, D_INNER=2048, D_STATE=16, L=1024
#define BSZ      2
#define DINNER   2048
#define DSTATE   16
#define LSEQ     1024
#define TILE     64             // timesteps per LDS tile (16 tiles over L)
#define SB_PITCH 68             // B tile pitch (floats): bank = 4n+i, conflict-free; 272B rows (16B aligned)
#define SH_PITCH 68             // output tile pitch: same properties

// One wave (32 lanes) per block; block owns (b, d-pair): lane = dd*16 + n.
// Per 64-step tile:
//   1) async-load delta/u (2x64) and B (16x64) tiles into LDS
//   2) serial scan: h = exp2(dv*(A*log2e))*h + dv*uv*bv, staged in sH[32][68]
//   3) async-store from LDS: 16 issues, each covering 2 output rows
//      (lanes 0-15 -> row 2r, lanes 16-31 -> row 2r+1), 2x256B contiguous.
// ~14 KB LDS/block -> ~22 single-wave workgroups per WGP for latency hiding.
__global__ __launch_bounds__(32)
void selective_scan_kernel(const float* __restrict__ u,
                           const float* __restrict__ delta,
                           const float* __restrict__ A,
                           const float* __restrict__ Bm,
                           const float* __restrict__ init_state,
                           float* __restrict__ out)
{
    __shared__ float sD[2 * TILE];
    __shared__ float sU[2 * TILE];
    __shared__ float sB[DSTATE * SB_PITCH];
    __shared__ float sH[32 * SH_PITCH];

    const int lane = threadIdx.x;        // 0..31
    const int dd   = lane >> 4;          // 0..1
    const int n    = lane & 15;          // 0..15
    const int b    = blockIdx.x >> 10;   // DINNER/2 = 1024 pairs per batch
    const int d0   = (blockIdx.x & 1023) << 1;
    const int d    = d0 + dd;

    // exp(x) == exp2(x * log2(e)) -> fold log2(e) into A once, one mul/step saved.
    const float AdnL2 = A[d * DSTATE + n] * 1.4426950408889634f;
    float h = init_state[d * DSTATE + n];

    const float* uRow = u     + ((size_t)b * DINNER + d0) * LSEQ;
    const float* dRow = delta + ((size_t)b * DINNER + d0) * LSEQ;
    const float* bRow = Bm    + (size_t)b * DSTATE * LSEQ;

    // Wave-relative LDS byte offsets (generic shared pointer low 32 bits).
    const uint32_t sD_base = (uint32_t)(uintptr_t)(&sD[0]);
    const uint32_t sU_base = (uint32_t)(uintptr_t)(&sU[0]);
    const uint32_t sB_base = (uint32_t)(uintptr_t)(&sB[0]);
    const uint32_t sH_base = (uint32_t)(uintptr_t)(&sH[0]);

    for (int t0 = 0; t0 < LSEQ; t0 += TILE) {
        // ---- Phase 1: async global->LDS loads (16B per lane per issue).
        // delta & u tiles: 2 rows x 64 floats = 32 b128 chunks each (1 issue).
        {
            int row = lane >> 4;             // 0..1
            int cir = lane & 15;             // chunk within row
            const float* gd = dRow + (size_t)row * LSEQ + t0 + cir * 4;
            const float* gu = uRow + (size_t)row * LSEQ + t0 + cir * 4;
            uint32_t ld = sD_base + (uint32_t)(row * TILE + cir * 4) * 4u;
            uint32_t lu = sU_base + (uint32_t)(row * TILE + cir * 4) * 4u;
            asm volatile("global_load_async_to_lds_b128 %0, %1, off th:TH_LOAD_NT"
                         :: "v"(ld), "v"(gd) : "memory");  // read-once stream
            asm volatile("global_load_async_to_lds_b128 %0, %1, off th:TH_LOAD_NT"
                         :: "v"(lu), "v"(gu) : "memory");
        }
        // B tile: 16 rows x 64 floats = 256 chunks (8 issues). Default (RT)
        // policy: B is only 128KB and re-read by all 2048 blocks from L2.
        #pragma unroll
        for (int k = 0; k < 8; ++k) {
            int c   = lane + 32 * k;         // 0..255
            int row = c >> 4;                // 0..15
            int cir = c & 15;
            const float* gb = bRow + (size_t)row * LSEQ + t0 + cir * 4;
            uint32_t lb = sB_base + (uint32_t)(row * SB_PITCH + cir * 4) * 4u;
            asm volatile("global_load_async_to_lds_b128 %0, %1, off"
                         :: "v"(lb), "v"(gb) : "memory");
        }
        // Drains this tile's loads AND the previous tile's stores (sH reuse).
        asm volatile("s_wait_asynccnt 0" ::: "memory");

        // ---- Phase 2: serial scan over the tile (parallel across 32 (d,n) lanes).
        #pragma unroll 4
        for (int i = 0; i < TILE; ++i) {
            float dv = sD[dd * TILE + i];        // broadcast to 16 lanes
            float uv = sU[dd * TILE + i];
            float bv = sB[n * SB_PITCH + i];     // conflict-free banks
            float a  = __builtin_amdgcn_exp2f(dv * AdnL2);   // v_exp_f32
            h = fmaf(a, h, dv * uv * bv);
            sH[lane * SH_PITCH + i] = h;         // staged for coalesced store
        }

        // Wave ds_stores must land before the async engine reads LDS.
        asm volatile("s_wait_dscnt 0" ::: "memory");

        // ---- Phase 3: async LDS->global stores, 2 rows per issue.
        // Output is a 256MB write-once stream (> L2): non-temporal.
        #pragma unroll
        for (int rr = 0; rr < 16; ++rr) {
            int row = 2 * rr + dd;               // lanes 0-15: even, 16-31: odd
            int col = (lane & 15) * 4;           // 4 floats per lane
            int nn  = row & 15;
            int rdd = row >> 4;
            float* go = out + ((((size_t)b * DSTATE + nn) * DINNER) + d0 + rdd) * LSEQ
                            + t0 + col;
            uint32_t lh = sH_base + (uint32_t)(row * SH_PITCH + col) * 4u;
            asm volatile("global_store_async_from_lds_b128 %0, %1, off th:TH_STORE_NT"
                         :: "v"(go), "v"(lh) : "memory");
        }
    }
    // Ensure final stores complete before wave exit (s_endpgm also wait-idles).
    asm volatile("s_wait_asynccnt 0" ::: "memory");
}

extern "C" void kernel_launch(void* const* d_in, const int* in_sizes, int n_in,
                              void* d_out, int out_size, void* d_ws, size_t ws_size,
                              hipStream_t stream) {
    (void)in_sizes; (void)n_in; (void)out_size; (void)d_ws; (void)ws_size;
    const float* u     = (const float*)d_in[0];
    const float* delta = (const float*)d_in[1];
    const float* A     = (const float*)d_in[2];
    const float* Bm    = (const float*)d_in[3];
    const float* initS = (const float*)d_in[4];
    float* out = (float*)d_out;

    dim3 grid(BSZ * DINNER / 2);   // 2048 single-wave workgroups
    dim3 block(32);
    selective_scan_kernel<<<grid, block, 0, stream>>>(u, delta, A, Bm, initS, out);
}